// PosedPointmapLoss_2602750181383
// MI455X (gfx1250) — compile-verified
//
#include <hip/hip_runtime.h>
#include <hip/hip_bf16.h>

// ---------------------------------------------------------------------------
// Chamfer loss on MI455X (gfx1250, wave32).
//
//   d2[m][n] = ||x_m||^2 - 2 x_m . y_n + ||y_n||^2
//
// The -2 x.y + ||x||^2 part is computed with V_WMMA_F32_16X16X4_F32 by packing
// A = [x0, x1, x2, ||x||^2] and B = [-2y0, -2y1, -2y2, 1] into the K=4 slots.
// One wave owns a 16-row tile, sweeps all 256 column tiles of its batch, and
// min-accumulates in 8 VGPRs (the 16x16 f32 C/D accumulator shape).
// ---------------------------------------------------------------------------

typedef float v2f __attribute__((ext_vector_type(2)));
typedef float v8f __attribute__((ext_vector_type(8)));

#define POINTS      4096
#define TILES       (POINTS / 16)     // 256 column tiles per sweep
#define BATCHES     8
#define TOTAL_ELEMS (BATCHES * POINTS)
#define BIG_F32     3.402823466e+38f

// One wave = one 16-row tile of X; sweep all columns of Y; write row-wise
// nearest-neighbor squared distances (exactly once, no atomics).
__global__ __launch_bounds__(256) void chamfer_rowmin_kernel(
    const float* __restrict__ X,   // (8, 4096, 3) query points (rows)
    const float* __restrict__ Y,   // (8, 4096, 3) target points (cols)
    float* __restrict__ out)       // (8, 4096) per-row min squared distance
{
    const int wave = (blockIdx.x * blockDim.x + threadIdx.x) >> 5;
    const int lane = threadIdx.x & 31;
    const int b    = wave >> 8;          // batch index (8 * 256 waves total)
    const int tile = wave & (TILES - 1); // row-tile within batch
    const int half = lane >> 4;          // 0: lanes 0-15, 1: lanes 16-31
    const int l16  = lane & 15;

    const float* __restrict__ Xb = X + (size_t)b * POINTS * 3;
    const float* __restrict__ Yb = Y + (size_t)b * POINTS * 3;

    // ---- A fragment: row M = l16; K layout per ISA 16x4 f32 A-matrix:
    //      lanes 0-15 hold K=0,1 in v0,v1; lanes 16-31 hold K=2,3.
    const int row = tile * 16 + l16;
    const float x0 = Xb[row * 3 + 0];
    const float x1 = Xb[row * 3 + 1];
    const float x2 = Xb[row * 3 + 2];
    const float xs = x0 * x0 + x1 * x1 + x2 * x2;
    v2f afrag;
    afrag.x = half ? x2 : x0;
    afrag.y = half ? xs : x1;

    // Row-min accumulators: acc[r] covers M = r + 8*half, N = l16 per tile.
    float acc[8];
#pragma unroll
    for (int r = 0; r < 8; ++r) acc[r] = BIG_F32;

    for (int t = 0; t < TILES; ++t) {
        // ---- B fragment: column N = l16 of this tile (both halves load the
        //      same column; half selects which K rows it supplies).
        const int col = t * 16 + l16;
        const float y0 = Yb[col * 3 + 0];
        const float y1 = Yb[col * 3 + 1];
        const float y2 = Yb[col * 3 + 2];
        const float ys = y0 * y0 + y1 * y1 + y2 * y2;
        v2f bfrag;
        bfrag.x = half ? (-2.0f * y2) : (-2.0f * y0);
        bfrag.y = half ? 1.0f         : (-2.0f * y1);

        v8f c = {0.f, 0.f, 0.f, 0.f, 0.f, 0.f, 0.f, 0.f};
        // D = ||x_m||^2 - 2 x_m . y_n   (one 16x16 tile per wave)
        c = __builtin_amdgcn_wmma_f32_16x16x4_f32(
                /*neg_a=*/false, afrag, /*neg_b=*/false, bfrag,
                /*c_mod=*/(short)0, c, /*reuse_a=*/false, /*reuse_b=*/false);

        // + ||y_n||^2 : N == l16 for every accumulator VGPR, so ys is the
        // right per-lane column term for all 8 rows. Then min-accumulate.
        // (max(0,.) clamp is deferred to the end: it commutes with min.)
#pragma unroll
        for (int r = 0; r < 8; ++r)
            acc[r] = fminf(acc[r], c[r] + ys);
    }

    // ---- Reduce over the 16 columns held by this half-wave (width-16
    //      butterflies keep lanes 0-15 / 16-31 separate, matching M=r vs r+8).
#pragma unroll
    for (int r = 0; r < 8; ++r) {
        float v = acc[r];
        v = fminf(v, __shfl_xor(v, 1, 16));
        v = fminf(v, __shfl_xor(v, 2, 16));
        v = fminf(v, __shfl_xor(v, 4, 16));
        v = fminf(v, __shfl_xor(v, 8, 16));
        acc[r] = v;
    }

    float* __restrict__ o = out + (size_t)b * POINTS + tile * 16;
    if (lane == 0) {
#pragma unroll
        for (int r = 0; r < 8; ++r) o[r] = fmaxf(acc[r], 0.0f);       // M = 0..7
    } else if (lane == 16) {
#pragma unroll
        for (int r = 0; r < 8; ++r) o[8 + r] = fmaxf(acc[r], 0.0f);   // M = 8..15
    }
}

__global__ void zero_out_kernel(float* out) { out[0] = 0.0f; }

// loss = mean(cham_x * mask) + mean(cham_y), both over 32768 elements.
__global__ __launch_bounds__(256) void reduce_loss_kernel(
    const float* __restrict__ cham_x,
    const float* __restrict__ cham_y,
    const float* __restrict__ mask,   // (2,4,1,64,64) flattens to (8,4096)
    float* __restrict__ out)
{
    float s = 0.0f;
    for (int i = blockIdx.x * blockDim.x + threadIdx.x; i < TOTAL_ELEMS;
         i += gridDim.x * blockDim.x) {
        s += cham_x[i] * mask[i] + cham_y[i];
    }
    // wave32 reduction
    s += __shfl_xor(s, 1, 32);
    s += __shfl_xor(s, 2, 32);
    s += __shfl_xor(s, 4, 32);
    s += __shfl_xor(s, 8, 32);
    s += __shfl_xor(s, 16, 32);

    __shared__ float smem[8];            // 256 threads -> 8 waves
    const int wave = threadIdx.x >> 5;
    const int lane = threadIdx.x & 31;
    if (lane == 0) smem[wave] = s;
    __syncthreads();
    if (wave == 0) {
        float t = (lane < 8) ? smem[lane] : 0.0f;
        t += __shfl_xor(t, 1, 32);
        t += __shfl_xor(t, 2, 32);
        t += __shfl_xor(t, 4, 32);
        if (lane == 0)
            atomicAdd(out, t * (1.0f / (float)TOTAL_ELEMS));
    }
}

extern "C" void kernel_launch(void* const* d_in, const int* in_sizes, int n_in,
                              void* d_out, int out_size, void* d_ws, size_t ws_size,
                              hipStream_t stream) {
    const float* v      = (const float*)d_in[0];  // (8, 4096, 3)
    const float* v_pred = (const float*)d_in[1];  // (8, 4096, 3)
    const float* mask   = (const float*)d_in[2];  // (2, 4, 1, 64, 64)
    float* out = (float*)d_out;

    float* cham_x = (float*)d_ws;                 // (8, 4096)
    float* cham_y = cham_x + TOTAL_ELEMS;         // (8, 4096)

    zero_out_kernel<<<1, 1, 0, stream>>>(out);

    // 8 batches * 256 row tiles = 2048 waves; 8 waves (256 thr) per block.
    const dim3 block(256);
    const dim3 grid((BATCHES * TILES * 32) / 256);

    // Pass 1: cham_x[p] = min_q ||v_pred_p - v_q||^2  (x = v_pred, y = v)
    chamfer_rowmin_kernel<<<grid, block, 0, stream>>>(v_pred, v, cham_x);
    // Pass 2: transposed matrix -> cham_y as row mins.
    chamfer_rowmin_kernel<<<grid, block, 0, stream>>>(v, v_pred, cham_y);

    reduce_loss_kernel<<<64, 256, 0, stream>>>(cham_x, cham_y, mask, out);
}